// OptimizedScaleShiftInvariantMACE_45861660787080
// MI455X (gfx1250) — compile-verified
//
#include <hip/hip_runtime.h>
#include <hip/hip_bf16.h>

typedef float v2f __attribute__((ext_vector_type(2)));
typedef float v8f __attribute__((ext_vector_type(8)));

#define C_DIM   64
#define NSH     16
#define NELEM_  10

// ---------------------------------------------------------------------------
// Zero the message accumulator (float4 stores, one store per thread)
// ---------------------------------------------------------------------------
__global__ __launch_bounds__(256) void zero_kernel(float4* __restrict__ p, size_t n4) {
    size_t i = (size_t)blockIdx.x * blockDim.x + threadIdx.x;
    size_t stride = (size_t)gridDim.x * blockDim.x;
    for (; i < n4; i += stride) p[i] = make_float4(0.f, 0.f, 0.f, 0.f);
}

// ---------------------------------------------------------------------------
// nf = node_feats @ W_up   via v_wmma_f32_16x16x4_f32
// one wave per 16-node tile; 4 d-tiles x 16 K-steps = 64 WMMA per wave
// ---------------------------------------------------------------------------
__global__ __launch_bounds__(256) void up_kernel(const float* __restrict__ x,
                                                 const float* __restrict__ W,
                                                 float* __restrict__ y, int N) {
    int wave = (blockIdx.x * blockDim.x + threadIdx.x) >> 5;   // wave-uniform
    int lane = threadIdx.x & 31;
    int tiles = N >> 4;
    if (wave >= tiles) return;                                  // uniform exit
    int mrow = lane & 15;
    int half = lane >> 4;
    int n0 = wave << 4;

    // A operand: rows = 16 consecutive nodes, preload 16 K-step v2f's
    v2f a[16];
    const float* ap = x + (size_t)(n0 + mrow) * C_DIM + 2 * half;
#pragma unroll
    for (int k = 0; k < 16; ++k) a[k] = *(const v2f*)(ap + 4 * k);

#pragma unroll 1
    for (int dt = 0; dt < 4; ++dt) {
        v8f acc = {};
        const float* Wp = W + dt * 16 + mrow;                   // + c*64
#pragma unroll
        for (int k = 0; k < 16; ++k) {
            int c0 = 4 * k + 2 * half;
            v2f b;
            b.x = Wp[(c0 + 0) * C_DIM];
            b.y = Wp[(c0 + 1) * C_DIM];
            acc = __builtin_amdgcn_wmma_f32_16x16x4_f32(
                false, a[k], false, b, (short)0, acc, false, false);
        }
        float* op = y + (size_t)(n0 + 8 * half) * C_DIM + dt * 16 + mrow;
#pragma unroll
        for (int j = 0; j < 8; ++j) op[(size_t)j * C_DIM] = acc[j];
    }
}

// ---------------------------------------------------------------------------
// W_comb[z][l] = W_lin[l] @ W_skip[z][l] / AVG_NEIGH      (40 tiny 64x64 GEMMs)
// one block of 64 threads per (z,l); thread owns output column d'
// ---------------------------------------------------------------------------
__global__ __launch_bounds__(64) void wcomb_kernel(const float* __restrict__ Wlin,
                                                   const float* __restrict__ Wskip,
                                                   float* __restrict__ Wc) {
    int zl = blockIdx.x;            // z*4 + l
    int l = zl & 3;
    int d2 = threadIdx.x;
    const float* A = Wlin + (size_t)l * C_DIM * C_DIM;          // [c][d]
    const float* B = Wskip + (size_t)zl * C_DIM * C_DIM;        // [d][d']
    float* O = Wc + (size_t)zl * C_DIM * C_DIM;

    float bcol[C_DIM];
#pragma unroll
    for (int d = 0; d < C_DIM; ++d) bcol[d] = B[(size_t)d * C_DIM + d2];

    for (int c = 0; c < C_DIM; ++c) {
        float acc = 0.f;
#pragma unroll
        for (int d = 0; d < C_DIM; ++d) acc += A[(size_t)c * C_DIM + d] * bcol[d];
        O[(size_t)c * C_DIM + d2] = acc * 0.05f;                // 1/AVG_NEIGH
    }
}

// ---------------------------------------------------------------------------
// Edge kernel: gather nf[sender], t_l = sf*ef_l, scatter sh_j * t_l(j) into
// msg[receiver] with L2 f32 atomics. 64 threads (one per channel) per edge.
// edge_feats streamed with non-temporal loads (read exactly once, 410 MB).
// ---------------------------------------------------------------------------
__global__ __launch_bounds__(256) void edge_kernel(const float* __restrict__ nf,
                                                   const float* __restrict__ edge_attrs,
                                                   const float* __restrict__ edge_feats,
                                                   const int* __restrict__ ei,
                                                   float* __restrict__ msg, int E) {
    long long t = (long long)blockIdx.x * blockDim.x + threadIdx.x;
    int e = (int)(t >> 6);
    int c = (int)(t & 63);
    if (e >= E) return;
    int s = ei[e];
    int r = ei[E + e];

    float sfc = nf[(size_t)s * C_DIM + c];
    const float* efp = edge_feats + (size_t)e * (4 * C_DIM) + c;
    float tl[4];
#pragma unroll
    for (int l = 0; l < 4; ++l)
        tl[l] = sfc * __builtin_nontemporal_load(efp + l * C_DIM);

    const float* shp = edge_attrs + (size_t)e * NSH;
    float* mp = msg + (size_t)r * (NSH * C_DIM) + c;
#pragma unroll
    for (int j = 0; j < NSH; ++j) {
        int l = (j >= 9) ? 3 : (j >= 4) ? 2 : (j >= 1) ? 1 : 0;
        unsafeAtomicAdd(mp + (size_t)j * C_DIM, shp[j] * tl[l]);
    }
}

// ---------------------------------------------------------------------------
// out[n] = sum_l maskrows_l(msg[n]) @ W_comb[elem(n)][l]
// one wave per node; A rows masked per l (rows l^2..(l+1)^2-1 valid), all four
// l's accumulate into the same 16x16 WMMA chain. 256 v_wmma per node.
// ---------------------------------------------------------------------------
__global__ __launch_bounds__(256) void node_out_kernel(const float* __restrict__ msg,
                                                       const float* __restrict__ Wc,
                                                       const float* __restrict__ node_attrs,
                                                       float* __restrict__ out, int N) {
    int wave = (blockIdx.x * blockDim.x + threadIdx.x) >> 5;   // wave-uniform
    int lane = threadIdx.x & 31;
    if (wave >= N) return;                                     // uniform exit
    int n = wave;
    int mrow = lane & 15;
    int half = lane >> 4;

    // element index from one-hot attrs (10 broadcast loads)
    int z = 0;
#pragma unroll
    for (int j = 0; j < NELEM_; ++j)
        if (node_attrs[(size_t)n * NELEM_ + j] > 0.5f) z = j;

    // preload A: this lane covers msg[n][mrow][c], c = 4k + 2*half (+0,+1)
    v2f a[16];
    const float* mp = msg + (size_t)n * (NSH * C_DIM) + (size_t)mrow * C_DIM + 2 * half;
#pragma unroll
    for (int k = 0; k < 16; ++k) a[k] = *(const v2f*)(mp + 4 * k);

    const float* Wz = Wc + (size_t)z * 4 * C_DIM * C_DIM;
    const v2f zero2 = {0.f, 0.f};

#pragma unroll 1
    for (int dt = 0; dt < 4; ++dt) {
        v8f acc = {};
#pragma unroll 1
        for (int l = 0; l < 4; ++l) {
            int lo = l * l, hi = (l + 1) * (l + 1);
            bool valid = (mrow >= lo) && (mrow < hi);
            const float* Wl = Wz + (size_t)l * C_DIM * C_DIM + dt * 16 + mrow;  // + c*64
#pragma unroll
            for (int k = 0; k < 16; ++k) {
                int c0 = 4 * k + 2 * half;
                v2f av = valid ? a[k] : zero2;
                v2f b;
                b.x = Wl[(c0 + 0) * C_DIM];
                b.y = Wl[(c0 + 1) * C_DIM];
                acc = __builtin_amdgcn_wmma_f32_16x16x4_f32(
                    false, av, false, b, (short)0, acc, false, false);
            }
        }
        float* op = out + (size_t)n * (NSH * C_DIM) + (size_t)(8 * half) * C_DIM + dt * 16 + mrow;
#pragma unroll
        for (int j = 0; j < 8; ++j) op[(size_t)j * C_DIM] = acc[j];
    }
}

// ---------------------------------------------------------------------------
extern "C" void kernel_launch(void* const* d_in, const int* in_sizes, int n_in,
                              void* d_out, int out_size, void* d_ws, size_t ws_size,
                              hipStream_t stream) {
    const float* node_attrs = (const float*)d_in[0];
    const float* node_feats = (const float*)d_in[1];
    const float* edge_attrs = (const float*)d_in[2];
    const float* edge_feats = (const float*)d_in[3];
    const float* W_up       = (const float*)d_in[4];
    const float* W_lin      = (const float*)d_in[5];
    const float* W_skip     = (const float*)d_in[6];
    const int*   edge_index = (const int*)d_in[7];

    int N = in_sizes[1] / C_DIM;           // 20000
    int E = in_sizes[3] / (4 * C_DIM);     // 400000

    // workspace layout (floats): nf[N*64] | Wc[40*4096] | msg[N*1024]
    float* nf  = (float*)d_ws;
    float* Wc  = nf + (size_t)N * C_DIM;
    float* msg = Wc + (size_t)NELEM_ * 4 * C_DIM * C_DIM;
    float* out = (float*)d_out;

    // 1) zero msg accumulator
    size_t n4 = (size_t)N * NSH * C_DIM / 4;
    int zb = (int)((n4 + 255) / 256);
    zero_kernel<<<zb, 256, 0, stream>>>((float4*)msg, n4);

    // 2) nf = node_feats @ W_up   (WMMA f32)
    int tiles = N / 16;                                  // 1250 waves
    up_kernel<<<(tiles * 32 + 255) / 256, 256, 0, stream>>>(node_feats, W_up, nf, N);

    // 3) W_comb precompute (folds W_lin, W_skip, 1/AVG_NEIGH)
    wcomb_kernel<<<NELEM_ * 4, 64, 0, stream>>>(W_lin, W_skip, Wc);

    // 4) edge gather / TP / atomic scatter
    long long ethreads = (long long)E * C_DIM;
    edge_kernel<<<(int)((ethreads + 255) / 256), 256, 0, stream>>>(
        nf, edge_attrs, edge_feats, edge_index, msg, E);

    // 5) per-node output GEMM with per-row-l matrix select (WMMA f32)
    node_out_kernel<<<(N * 32 + 255) / 256, 256, 0, stream>>>(msg, Wc, node_attrs, out, N);
}